// StackedBladeBank_8186207666948
// MI455X (gfx1250) — compile-verified
//
#include <hip/hip_runtime.h>

// StackedBladeBank: FNV-1 hash -> 8-blade gather -> streamed transpose-out.
// out[w][blade][0:32] = bank[blade][fnv1(byte_window[w]) % N_SLOTS][0:32]
//
// Memory-bound: ~264 MB/call => ~11 us at 23.3 TB/s. No matmul FLOPs, so no
// WMMA. CDNA5-specific shape of this kernel:
//   - 1 wave32 == 1 window: window index forced wave-uniform with
//     readfirstlane, so the 64B window fetch goes through the scalar path
//     (s_load / constant cache, KMcnt) and the whole FNV chain + modulo runs
//     on the SALU, co-issuing beside the vector memory stream.
//   - gather loads are GVS-form (SGPR base+addr, small per-lane voffset),
//     8 lanes x b128 == exactly one 128B bank row per cacheline.
//   - output is write-once => global_store_b128 th:TH_STORE_NT, preserving
//     the 192MB L2 for the 512MB bank's reuse.

#define N_GRAM   16
#define D_STATE  32
#define N_BLADES 8

typedef float __attribute__((ext_vector_type(4))) f32x4;
typedef int   __attribute__((ext_vector_type(4))) i32x4;

// NSLOTS_CT > 0: compile-time slot count (modulo folds to mulhi sequence).
// NSLOTS_CT == 0: generic runtime-divisor fallback.
template <int NSLOTS_CT>
__global__ __launch_bounds__(256) void blade_gather_kernel(
    const int* __restrict__ byte_window,   // [n_windows][16] int32 (0..255)
    const float* __restrict__ bank,        // [8][n_slots][32] f32
    float* __restrict__ out,               // [n_windows][8][32] f32
    unsigned n_windows,
    unsigned n_slots_rt)
{
    const unsigned n_slots = (NSLOTS_CT > 0) ? (unsigned)NSLOTS_CT : n_slots_rt;
    const unsigned lane = threadIdx.x & 31u;

    // One wave per window; 8 windows per 256-thread block. Force the window
    // index wave-uniform so the hash scalarizes (SMEM load + SALU math).
    unsigned w = blockIdx.x * 8u + (threadIdx.x >> 5);
    w = (unsigned)__builtin_amdgcn_readfirstlane((int)w);
    if (w >= n_windows) return;            // uniform branch, no barrier in kernel

    // FNV-1 over the window's 16 bytes (uniform address -> scalar loads).
    const i32x4* bw = (const i32x4*)(byte_window + (size_t)w * N_GRAM);
    unsigned h = 2166136261u;
#pragma unroll
    for (int v = 0; v < 4; ++v) {
        i32x4 q = bw[v];
        h = (h ^ (unsigned)q.x) * 16777619u;
        h = (h ^ (unsigned)q.y) * 16777619u;
        h = (h ^ (unsigned)q.z) * 16777619u;
        h = (h ^ (unsigned)q.w) * 16777619u;
    }
    const unsigned addr = h % n_slots;     // uniform (SALU mulhi for CT slots)

    // Lane -> (blade, float4-part). Lane l covers output chunks l and l+32:
    // chunks 0..31 = blades 0..3, chunks 32..63 = blades 4..7, same addr.
    const unsigned blade_lo = lane >> 3;   // 0..3
    const unsigned part     = lane & 7u;   // float4 index within 32-float row

    const f32x4* bankv = (const f32x4*)bank;
    const size_t row_f4   = (size_t)(D_STATE / 4);              // 8
    const size_t blade_f4 = (size_t)n_slots * row_f4;           // per-blade stride
    const size_t base     = ((size_t)blade_lo * n_slots + addr) * row_f4 + part;

    // Two gathers per lane: 8 consecutive lanes read one contiguous 128B row.
    const f32x4 v0 = bankv[base];                               // blade_lo
    const f32x4 v1 = bankv[base + 4u * blade_f4];               // blade_lo + 4

    // Streamed, fully coalesced, write-once output -> non-temporal b128 stores.
    f32x4* dstv = (f32x4*)out + (size_t)w * 64u;                // 64 f32x4 / window
    __builtin_nontemporal_store(v0, dstv + lane);
    __builtin_nontemporal_store(v1, dstv + lane + 32u);
}

extern "C" void kernel_launch(void* const* d_in, const int* in_sizes, int n_in,
                              void* d_out, int out_size, void* d_ws, size_t ws_size,
                              hipStream_t stream)
{
    (void)n_in; (void)out_size; (void)d_ws; (void)ws_size;

    const int*   byte_window = (const int*)d_in[0];    // int32
    const float* bank        = (const float*)d_in[1];  // float32

    const unsigned n_windows = (unsigned)(in_sizes[0] / N_GRAM);               // 131072
    const unsigned n_slots   = (unsigned)(in_sizes[1] / (N_BLADES * D_STATE)); // 500000

    // 8 windows (waves) per 256-thread block.
    dim3 grid((n_windows + 7u) / 8u), block(256);

    if (n_slots == 500000u) {
        blade_gather_kernel<500000><<<grid, block, 0, stream>>>(
            byte_window, bank, (float*)d_out, n_windows, n_slots);
    } else {
        blade_gather_kernel<0><<<grid, block, 0, stream>>>(
            byte_window, bank, (float*)d_out, n_windows, n_slots);
    }
}